// ContrastiveBoundaryLoss_72310069395902
// MI455X (gfx1250) — compile-verified
//
#include <hip/hip_runtime.h>
#include <hip/hip_bf16.h>
#include <math.h>

// ContrastiveBoundaryLoss for MI455X (gfx1250, wave32).
// Memory-bound scatter/segment-sum workload. Traffic budget:
//   pass1: idx_i+idx_j streams 16B/edge = 102.4 MB (+6.4 MB class write)
//   pass2: idx_i 8B + logit 4B + class 1B per edge = 83.2 MB
//   total ~198 MB -> ~8.5 us floor at 23.3 TB/s.
// CDNA5 paths: async global->LDS staging (ASYNCcnt + s_wait_asynccnt) of the
// contiguous logit/class streams; random label/flag gathers ride in L2 (192MB).

#define CBL_THREADS 256      // 8 wave32 per block
#define CBL_TILE    1024     // edges staged per block in pass 2
#define CBL_INV_T   (1.0f / 0.07f)
#define CBL_EPS     1e-8f

#if defined(__gfx1250__) && __has_builtin(__builtin_amdgcn_global_load_async_to_lds_b32)
#define CBL_ASYNC_LDS 1
#define AS1 __attribute__((address_space(1)))
#define AS3 __attribute__((address_space(3)))
#endif

__device__ __forceinline__ void cbl_async_b32(const void* g, void* lds) {
#if defined(CBL_ASYNC_LDS)
    __builtin_amdgcn_global_load_async_to_lds_b32(
        (AS1 int*)(unsigned long long)g,
        (AS3 int*)(AS3 void*)lds,
        /*imm offset*/ 0, /*cpol*/ 0);
#else
    *(int*)lds = *(const int*)g;   // host-pass / non-gfx1250 fallback
#endif
}

__device__ __forceinline__ void cbl_async_wait() {
#if defined(CBL_ASYNC_LDS)
    asm volatile("s_wait_asynccnt 0x0" ::: "memory");
#endif
}

// ---------------------------------------------------------------- init ------
// Compress int64 labels to int32 (random-gather table: 400 KB, L2 resident)
// and zero the per-node accumulators.
__global__ void cbl_init(const long long* __restrict__ label,
                         int* __restrict__ lab32,
                         int* __restrict__ bflag,
                         float* __restrict__ denom,
                         float* __restrict__ numer,
                         float* __restrict__ accum,
                         int N) {
    int i = blockIdx.x * blockDim.x + threadIdx.x;
    if (i < N) {
        lab32[i] = (int)label[i];
        bflag[i] = 0;
        denom[i] = 0.0f;
        numer[i] = 0.0f;
    }
    if (i == 0) { accum[0] = 0.0f; accum[1] = 0.0f; }
}

// ------------------------------------------------------- pass 1 (fast) ------
// 2 edges per thread via b128 index loads. Emits per-edge class byte:
//   0 = invalid, 1 = valid negative, 2 = valid positive.
// Boundary flag: node has >=1 valid negative outgoing edge.
__global__ void cbl_pass1_fast(const int4* __restrict__ ii4,
                               const int4* __restrict__ jj4,
                               const int* __restrict__ lab32,
                               int* __restrict__ bflag,
                               unsigned short* __restrict__ ecl16,
                               int Epair) {
    int t = blockIdx.x * blockDim.x + threadIdx.x;
    if (t >= Epair) return;
    int4 a = ii4[t];              // {i0.lo, i0.hi, i1.lo, i1.hi}
    int4 b = jj4[t];
    int i0 = a.x, i1 = a.z;
    int j0 = b.x, j1 = b.z;
    int li0 = lab32[i0], lj0 = lab32[j0];
    int li1 = lab32[i1], lj1 = lab32[j1];
    unsigned int c0 = (li0 != -1 && lj0 != -1) ? ((li0 == lj0) ? 2u : 1u) : 0u;
    unsigned int c1 = (li1 != -1 && lj1 != -1) ? ((li1 == lj1) ? 2u : 1u) : 0u;
    ecl16[t] = (unsigned short)(c0 | (c1 << 8));
    if (c0 == 1u) atomicOr(&bflag[i0], 1);   // fire-and-forget (STOREcnt)
    if (c1 == 1u) atomicOr(&bflag[i1], 1);
}

// ------------------------------------------------------- pass 1 (slow) ------
__global__ void cbl_pass1(const int2* __restrict__ idx_i,
                          const int2* __restrict__ idx_j,
                          const int* __restrict__ lab32,
                          int* __restrict__ bflag,
                          int E) {
    int e = blockIdx.x * blockDim.x + threadIdx.x;
    if (e >= E) return;
    int i = idx_i[e].x;
    int j = idx_j[e].x;
    int li = lab32[i];
    int lj = lab32[j];
    if (li != -1 && lj != -1 && li != lj) atomicOr(&bflag[i], 1);
}

// ------------------------------------------------------- pass 2 (fast) ------
// Async-stage the logit tile (4 KB) and class tile (1 KB) into LDS, then each
// thread processes 4 consecutive edges with b128 index loads and b128 LDS
// reads. Requires E % 4 == 0 (host-gated).
__global__ void cbl_pass2_fast(const int* __restrict__ rawii,   // 2*E dwords
                               const float* __restrict__ logits,
                               const unsigned char* __restrict__ eclass,
                               const int* __restrict__ bflag,
                               float* __restrict__ denom,
                               float* __restrict__ numer,
                               int E) {
    __shared__ float slog[CBL_TILE];
    __shared__ unsigned char scls[CBL_TILE];
    const int base = blockIdx.x * CBL_TILE;
    const int tid  = threadIdx.x;

    for (int k = tid; k < CBL_TILE; k += CBL_THREADS) {
        int e = base + k;
        if (e < E) cbl_async_b32(logits + e, &slog[k]);
    }
    {   // CBL_TILE/4 == CBL_THREADS dwords of class bytes
        int d = (base >> 2) + tid;
        if (4 * d < E) cbl_async_b32((const int*)eclass + d, &scls[4 * tid]);
    }
    cbl_async_wait();
    __syncthreads();

    const int k0 = tid * 4;
    const int e0 = base + k0;
    if (e0 >= E) return;                       // E%4==0 => e0+3 < E here
    const int4* p = (const int4*)(rawii + 2 * e0);
    int4 a = p[0], b = p[1];
    int idx[4] = {a.x, a.z, b.x, b.z};
    float4 lg = *(const float4*)&slog[k0];
    float lgv[4] = {lg.x, lg.y, lg.z, lg.w};
    unsigned int cw = *(const unsigned int*)&scls[k0];
#pragma unroll
    for (int u = 0; u < 4; ++u) {
        unsigned int c = (cw >> (8 * u)) & 0xffu;
        if (c != 0u) {
            int i = idx[u];
            if (bflag[i] != 0) {
                float w = expf(lgv[u] * CBL_INV_T);
                atomicAdd(&denom[i], w);
                if (c == 2u) atomicAdd(&numer[i], w);
            }
        }
    }
}

// ------------------------------------------------------- pass 2 (slow) ------
__global__ void cbl_pass2(const int2* __restrict__ idx_i,
                          const int2* __restrict__ idx_j,
                          const float* __restrict__ logits,
                          const int* __restrict__ lab32,
                          const int* __restrict__ bflag,
                          float* __restrict__ denom,
                          float* __restrict__ numer,
                          int E) {
    __shared__ float slog[CBL_TILE];
    const int base = blockIdx.x * CBL_TILE;
    const int tid  = threadIdx.x;
    for (int k = tid; k < CBL_TILE; k += CBL_THREADS) {
        int e = base + k;
        if (e < E) cbl_async_b32(logits + e, &slog[k]);
    }
    cbl_async_wait();
    __syncthreads();
    for (int k = tid; k < CBL_TILE; k += CBL_THREADS) {
        int e = base + k;
        if (e >= E) break;
        int i = idx_i[e].x;
        int j = idx_j[e].x;
        int li = lab32[i];
        int lj = lab32[j];
        if (li != -1 && lj != -1 && bflag[i] != 0) {
            float w = expf(slog[k] * CBL_INV_T);
            atomicAdd(&denom[i], w);
            if (li == lj) atomicAdd(&numer[i], w);
        }
    }
}

// ---------------------------------------------------------------- nodes -----
__global__ void cbl_nodes(const int* __restrict__ bflag,
                          const float* __restrict__ denom,
                          const float* __restrict__ numer,
                          float* __restrict__ accum,
                          int N) {
    __shared__ float ssum[CBL_THREADS];
    __shared__ float scnt[CBL_THREADS];
    int n = blockIdx.x * blockDim.x + threadIdx.x;
    float per = 0.0f, cnt = 0.0f;
    if (n < N) {
        float d = denom[n];
        if (bflag[n] != 0 && d > 0.0f) {   // bflag already implies label != -1
            per = logf(d + CBL_EPS) - logf(numer[n] + CBL_EPS);
            cnt = 1.0f;
        }
    }
    ssum[threadIdx.x] = per;
    scnt[threadIdx.x] = cnt;
    __syncthreads();
    for (int s = CBL_THREADS / 2; s > 0; s >>= 1) {
        if (threadIdx.x < s) {
            ssum[threadIdx.x] += ssum[threadIdx.x + s];
            scnt[threadIdx.x] += scnt[threadIdx.x + s];
        }
        __syncthreads();
    }
    if (threadIdx.x == 0) {
        atomicAdd(&accum[0], ssum[0]);
        atomicAdd(&accum[1], scnt[0]);
    }
}

// ---------------------------------------------------------------- final -----
__global__ void cbl_final(const float* __restrict__ accum,
                          float* __restrict__ out) {
    out[0] = accum[0] / fmaxf(accum[1], 1.0f);
}

// ---------------------------------------------------------------- launch ----
extern "C" void kernel_launch(void* const* d_in, const int* in_sizes, int n_in,
                              void* d_out, int out_size, void* d_ws, size_t ws_size,
                              hipStream_t stream) {
    const long long* edge_index = (const long long*)d_in[0]; // (2, E) int64
    const float*     logits     = (const float*)d_in[1];     // (E,)  f32
    const long long* label      = (const long long*)d_in[2]; // (N,)  int64
    const int E = in_sizes[1];
    const int N = in_sizes[2];

    char* ws = (char*)d_ws;
    int*   lab32 = (int*)(ws);
    int*   bflag = (int*)(ws + (size_t)4  * N);
    float* denom = (float*)(ws + (size_t)8  * N);
    float* numer = (float*)(ws + (size_t)12 * N);
    float* accum = (float*)(ws + (size_t)16 * N);
    unsigned char* eclass = (unsigned char*)(ws + (size_t)16 * N + 16);

    const size_t need_fast = (size_t)16 * N + 16 + (size_t)((E + 3) & ~3);
    const bool fast = (E % 4 == 0) && (ws_size >= need_fast);

    const int2* ii = (const int2*)edge_index;        // low dwords of idx_i
    const int2* jj = (const int2*)(edge_index + E);  // low dwords of idx_j

    cbl_init<<<(N + CBL_THREADS - 1) / CBL_THREADS, CBL_THREADS, 0, stream>>>(
        label, lab32, bflag, denom, numer, accum, N);

    if (fast) {
        const int Epair = E / 2;
        cbl_pass1_fast<<<(Epair + CBL_THREADS - 1) / CBL_THREADS, CBL_THREADS, 0, stream>>>(
            (const int4*)edge_index, (const int4*)(edge_index + E),
            lab32, bflag, (unsigned short*)eclass, Epair);
        cbl_pass2_fast<<<(E + CBL_TILE - 1) / CBL_TILE, CBL_THREADS, 0, stream>>>(
            (const int*)edge_index, logits, eclass, bflag, denom, numer, E);
    } else {
        cbl_pass1<<<(E + CBL_THREADS - 1) / CBL_THREADS, CBL_THREADS, 0, stream>>>(
            ii, jj, lab32, bflag, E);
        cbl_pass2<<<(E + CBL_TILE - 1) / CBL_TILE, CBL_THREADS, 0, stream>>>(
            ii, jj, logits, lab32, bflag, denom, numer, E);
    }

    cbl_nodes<<<(N + CBL_THREADS - 1) / CBL_THREADS, CBL_THREADS, 0, stream>>>(
        bflag, denom, numer, accum, N);

    cbl_final<<<1, 1, 0, stream>>>(accum, (float*)d_out);
}